// Window_Crop_53858889892321
// MI455X (gfx1250) — compile-verified
//
#include <hip/hip_runtime.h>
#include <stdint.h>

typedef __attribute__((ext_vector_type(2))) float    v2f;
typedef __attribute__((ext_vector_type(8))) float    v8f;
typedef __attribute__((ext_vector_type(4))) unsigned u32x4;
typedef __attribute__((ext_vector_type(8))) int      i32x8;
typedef __attribute__((ext_vector_type(4))) int      i32x4;

#define HH       112
#define NPIX     12544          // 112*112
#define SW       128            // SAT row stride in floats (bank-friendly)
#define SATN     (113 * SW)
#define NT       256
#define TOTAL_W  32459
// group offsets: 0, 6561, 12985, 19409, 25934 ; W1 = 25934

// ---- per-ratio window scoring via SAT lookups (constexpr dims => magic div)
template <int KH, int KW, int OH, int OW, int OFF, bool TRACK>
__device__ __forceinline__ void windows(const float* __restrict__ sat,
                                        float* __restrict__ out,
                                        float& bestV, int& bestI) {
  constexpr float inv = 1.0f / (float)(KH * KW);
  constexpr int total = OH * OW;
  for (int w = threadIdx.x; w < total; w += NT) {
    int y  = w / OW;
    int xx = w - y * OW;
    const float* s0 = sat + y * SW + xx;
    float s = s0[KH * SW + KW] - s0[KW] - s0[KH * SW] + s0[0];
    float v = s * inv;
    __builtin_nontemporal_store(v, out + OFF + w);
    if (TRACK) {
      if (v > bestV) { bestV = v; bestI = OFF + w; }
    }
  }
}

__global__ __launch_bounds__(NT) void window_score_kernel(
    const float* __restrict__ x, float* __restrict__ outIdx,
    float* __restrict__ outScore, float* __restrict__ outAll) {
  extern __shared__ float smem[];
  float* raw = smem;            // 12544 floats: this block's image
  float* sat = smem + NPIX;     // 113 x 128 floats: padded SAT

  const int tid = threadIdx.x;
  const int b   = blockIdx.x;

  // ---- 1) TDM: DMA this image (50 KB) global -> LDS (one wave issues)
  if (tid == 0) {
    uint64_t ga     = (uint64_t)(uintptr_t)(x + (size_t)b * NPIX);
    unsigned ldsOff = (unsigned)(uintptr_t)(void*)raw;  // low 32 bits = LDS byte offset
    u32x4 g0;
    g0[0] = 1u;                                         // count=1, user D#
    g0[1] = ldsOff;                                     // lds_addr
    g0[2] = (unsigned)ga;                               // global_addr[31:0]
    g0[3] = (unsigned)((ga >> 32) & 0x1FFFFFFu) | (2u << 30);  // addr[56:32] | type=2
    i32x8 g1;
    g1[0] = 0x00020000;                                 // workgroup_mask=0, data_size=2 (4B)
    g1[1] = (int)(((unsigned)NPIX & 0xFFFFu) << 16);    // tensor_dim0 lo16
    g1[2] = (int)(1u << 16);                            // tensor_dim0 hi=0, tensor_dim1=1
    g1[3] = (int)(((unsigned)NPIX & 0xFFFFu) << 16);    // tile_dim0 = 12544
    g1[4] = 0;                                          // tile_dim1=0, tile_dim2=0 (1-D)
    g1[5] = NPIX;                                       // tensor_dim0_stride lo32
    g1[6] = 0;
    g1[7] = 0;
    i32x4 gz; gz[0] = gz[1] = gz[2] = gz[3] = 0;
#if __clang_major__ >= 23
    i32x8 gz8; gz8[0]=gz8[1]=gz8[2]=gz8[3]=gz8[4]=gz8[5]=gz8[6]=gz8[7]=0;
    __builtin_amdgcn_tensor_load_to_lds(g0, g1, gz, gz, gz8, 0);
#else
    __builtin_amdgcn_tensor_load_to_lds(g0, g1, gz, gz, 0);
#endif
  }
  // zero the SAT (row 0 / col 0 padding + junk cols) while the TDM streams
  for (int i = tid; i < SATN; i += NT) sat[i] = 0.0f;
  if (tid == 0) __builtin_amdgcn_s_wait_tensorcnt(0);
  __syncthreads();

  // ---- 2) Row-direction prefix sums as triangular WMMA matmul: RP = X * U
  //      U[k][j] = (k<=j). Wave w owns row-strip mi=w with 7 independent
  //      accumulators (one per column tile). Each A chunk is loaded once and
  //      reused by up to 7 back-to-back WMMAs; below-diagonal tiles use a
  //      shared all-ones B register; above-diagonal chunks statically skipped.
  {
    const int lane = tid & 31;
    const int wave = tid >> 5;
    const int lm   = lane & 15;
    const int hi   = lane >> 4;                 // 0: K pair {0,1}, 1: K pair {2,3}
    if (wave < 7) {
      const int m0   = wave * 16;
      const int arow = (m0 + lm) * HH;
      v8f acc[7];
#pragma unroll
      for (int j = 0; j < 7; ++j) acc[j] = (v8f){};
      v2f bone;
      bone[0] = 1.0f; bone[1] = 1.0f;
#pragma unroll
      for (int s = 0; s < 28; ++s) {            // K chunks of 4
        const int k0 = s * 4 + hi * 2;
        v2f a;
        a[0] = raw[arow + k0];                  // A tile per ISA 16x4 f32 layout
        a[1] = raw[arow + k0 + 1];
        const int njmin = s >> 2;               // boundary (diagonal) column tile
        {
          const int jB = (njmin << 4) + lm;
          v2f bt;
          bt[0] = (k0     <= jB) ? 1.0f : 0.0f;
          bt[1] = (k0 + 1 <= jB) ? 1.0f : 0.0f;
          acc[njmin] = __builtin_amdgcn_wmma_f32_16x16x4_f32(
              false, a, false, bt, (short)0, acc[njmin], false, false);
        }
#pragma unroll
        for (int j = 0; j < 7; ++j) {
          if (j > njmin) {                      // fully below diagonal: B == ones
            acc[j] = __builtin_amdgcn_wmma_f32_16x16x4_f32(
                false, a, false, bone, (short)0, acc[j], false, false);
          }
        }
      }
      // store C tiles: VGPR v -> rows m0+v (lanes 0-15) / m0+v+8 (lanes 16-31)
      const int rbase = m0 + hi * 8 + 1;        // +1: SAT pad row
#pragma unroll
      for (int j = 0; j < 7; ++j) {
        const int cc = (j << 4) + lm + 1;       // +1: SAT pad column
#pragma unroll
        for (int v = 0; v < 8; ++v) sat[(rbase + v) * SW + cc] = acc[j][v];
      }
    }
  }
  __syncthreads();

  // ---- 3) Column-direction prefix (in place, one thread per column)
  if (tid < 113) {
    float carry = 0.0f;
    for (int r = 1; r <= HH; ++r) {
      carry += sat[r * SW + tid];
      sat[r * SW + tid] = carry;
    }
  }
  __syncthreads();

  // ---- 4) Score every window of every ratio; track argmax over first 4 groups
  float bestV = -__builtin_huge_valf();
  int   bestI = 0;
  float* outB = outAll + (size_t)b * TOTAL_W;
  windows<32, 32, 81, 81, 0,     true >(sat, outB, bestV, bestI);
  windows<25, 40, 88, 73, 6561,  true >(sat, outB, bestV, bestI);
  windows<40, 25, 73, 88, 12985, true >(sat, outB, bestV, bestI);
  windows<38, 26, 75, 87, 19409, true >(sat, outB, bestV, bestI);
  windows<26, 38, 87, 75, 25934, false>(sat, outB, bestV, bestI);

  // ---- 5) Block argmax reduction (first-index tie-break, matches jnp.argmax)
  __shared__ float redV[NT];
  __shared__ int   redI[NT];
  redV[tid] = bestV;
  redI[tid] = bestI;
  __syncthreads();
  for (int s = NT / 2; s > 0; s >>= 1) {
    if (tid < s) {
      float v2 = redV[tid + s]; int i2 = redI[tid + s];
      float v1 = redV[tid];     int i1 = redI[tid];
      if (v2 > v1 || (v2 == v1 && i2 < i1)) { redV[tid] = v2; redI[tid] = i2; }
    }
    __syncthreads();
  }
  if (tid == 0) {
    outIdx[b]   = (float)redI[0];
    outScore[b] = redV[0];
  }
}

extern "C" void kernel_launch(void* const* d_in, const int* in_sizes, int n_in,
                              void* d_out, int out_size, void* d_ws, size_t ws_size,
                              hipStream_t stream) {
  const float* x = (const float*)d_in[0];
  const int B = in_sizes[0] / NPIX;            // 1024
  float* out      = (float*)d_out;
  float* outIdx   = out;                       // (B,1) argmax indices (as float)
  float* outScore = out + B;                   // (B,1) proposal scores
  float* outAll   = out + 2 * (size_t)B;       // (B,32459) all window scores
  size_t shmem = (size_t)(NPIX + SATN) * sizeof(float);  // ~108 KB
  window_score_kernel<<<B, NT, shmem, stream>>>(x, outIdx, outScore, outAll);
}